// ENIGMAFinetune_45071386804438
// MI455X (gfx1250) — compile-verified
//
#include <hip/hip_runtime.h>
#include <hip/hip_bf16.h>
#include <math.h>

// ---------------------------------------------------------------------------
// MI455X (gfx1250) implementation. All GEMMs use V_WMMA_F32_16X16X4_F32
// (exact fp32 matrix MACs). One wave32 computes a 16x32 output strip
// (two 16x16 WMMA tiles sharing the A fragment, two accumulator chains).
//
// Fragment layouts (CDNA5 ISA 7.12.2):
//   A (16x4, MxK): lanes 0-15  -> row M=lane,    v.x=K(k),   v.y=K(k+1)
//                  lanes 16-31 -> row M=lane-16, v.x=K(k+2), v.y=K(k+3)
//   B (4x16, KxN): lanes 0-15  -> col N=lane,    v.x=K(k),   v.y=K(k+1)
//                  lanes 16-31 -> col N=lane-16, v.x=K(k+2), v.y=K(k+3)
//   C/D (16x16):   VGPR r: lanes 0-15 -> M=r, N=lane; lanes 16-31 -> M=8+r
// ---------------------------------------------------------------------------

typedef __attribute__((ext_vector_type(2))) float v2f;
typedef __attribute__((ext_vector_type(8))) float v8f;

#define B_N    1024
#define C_N    64
#define T_N    275
#define CT_N   (C_N * T_N)   // 17600
#define EMB_N  184
#define PROJ_N 1024

__device__ __forceinline__ v8f wmma4(v2f a, v2f b, v8f c) {
  return __builtin_amdgcn_wmma_f32_16x16x4_f32(false, a, false, b, (short)0, c,
                                               false, false);
}

__device__ __forceinline__ float gelu_exact(float x) {
  return 0.5f * x * (1.0f + erff(x * 0.70710678118654752440f));
}

// ---------------------------------------------------------------------------
// Kernel 1: subject-wise alignment.
// x[b, c, u] = sum_t eeg[b,c,t] * subj_W[sid[b], u, t] + subj_b[sid[b], u]
// Per batch: (64 x 275) @ (275 x 275)^T.
// Strips: 4 (M) x 9 (N/32) = 36 waves/batch. grid = (9, 1024), block = 128.
// Main K-loop is guard-free (K=0..271); branchless tail covers K=272..274.
// ---------------------------------------------------------------------------
__global__ __launch_bounds__(128) void subj_gemm_kernel(
    const float* __restrict__ eeg, const int* __restrict__ sid,
    const float* __restrict__ sW, const float* __restrict__ sB,
    float* __restrict__ x) {
  const int b    = blockIdx.y;
  const int wave = threadIdx.x >> 5;
  const int lane = threadIdx.x & 31;
  const int tile = blockIdx.x * 4 + wave;   // 0..35
  const int mt = tile / 9, np = tile % 9;
  const int m0 = mt * 16, n0 = np * 32;
  const int half = lane >> 4;
  const int l    = lane & 15;

  const int s = sid[b];
  const float* __restrict__ A    = eeg + (size_t)b * CT_N;      // [64,275]
  const float* __restrict__ W    = sW + (size_t)s * T_N * T_N;  // [275,275]
  const float* __restrict__ bias = sB + (size_t)s * T_N;

  const int  arow  = m0 + l;                 // < 64 always
  const int  ncol0 = n0 + l;                 // <= 271, always valid
  const int  ncol1 = n0 + 16 + l;            // may be >= 275 (last strip)
  const bool n1ok  = (ncol1 < T_N);
  const int  nc1   = n1ok ? ncol1 : 0;       // clamped; its cols never stored

  const float* __restrict__ Arp = A + (size_t)arow * T_N;
  const float* __restrict__ B0p = W + (size_t)ncol0 * T_N;
  const float* __restrict__ B1p = W + (size_t)nc1 * T_N;

  v8f acc0 = {}, acc1 = {};
  for (int k = 0; k < 272; k += 8) {        // 34 iters, no bounds checks
    const int ka = k + 2 * half;
    v2f a0, a1, b00, b01, b10, b11;
    a0.x  = Arp[ka];     a0.y  = Arp[ka + 1];
    a1.x  = Arp[ka + 4]; a1.y  = Arp[ka + 5];
    b00.x = B0p[ka];     b00.y = B0p[ka + 1];
    b01.x = B0p[ka + 4]; b01.y = B0p[ka + 5];
    b10.x = B1p[ka];     b10.y = B1p[ka + 1];
    b11.x = B1p[ka + 4]; b11.y = B1p[ka + 5];
    acc0 = wmma4(a0, b00, acc0);
    acc1 = wmma4(a0, b10, acc1);
    acc0 = wmma4(a1, b01, acc0);
    acc1 = wmma4(a1, b11, acc1);
  }
  {
    // Tail K=272..274: half=0 lanes carry K=272,273; half=1 lanes K=274 + 0.
    // Branchless: .y loads a valid in-bounds float and is masked to exact 0.
    const int   ka    = 272 + 2 * half;      // 272 or 274 (both in-bounds)
    const float ymask = half ? 0.0f : 1.0f;
    v2f a, b0, b1;
    a.x  = Arp[ka];  a.y  = Arp[273] * ymask;
    b0.x = B0p[ka];  b0.y = B0p[273] * ymask;
    b1.x = B1p[ka];  b1.y = B1p[273] * ymask;
    acc0 = wmma4(a, b0, acc0);
    acc1 = wmma4(a, b1, acc1);
  }

  float* __restrict__ X = x + (size_t)b * CT_N;
  const float bn0 = bias[ncol0];
#pragma unroll
  for (int r = 0; r < 8; ++r) {
    const int row = m0 + r + 8 * half;
    X[row * T_N + ncol0] = acc0[r] + bn0;
  }
  if (n1ok) {
    const float bn1 = bias[ncol1];
#pragma unroll
    for (int r = 0; r < 8; ++r) {
      const int row = m0 + r + 8 * half;
      X[row * T_N + ncol1] = acc1[r] + bn1;
    }
  }
}

// ---------------------------------------------------------------------------
// Kernel 2: encoder stub. h = x.reshape(1024, 17600) @ enc_W + enc_b
// M=1024 (64 tiles), N=184 (6 strips of 32), K=17600.
// 384 waves -> 48 blocks x 8 waves.
// ---------------------------------------------------------------------------
__global__ __launch_bounds__(256) void enc_gemm_kernel(
    const float* __restrict__ x, const float* __restrict__ eW,
    const float* __restrict__ eB, float* __restrict__ h) {
  const int wave = threadIdx.x >> 5;
  const int lane = threadIdx.x & 31;
  const int tile = blockIdx.x * 8 + wave;   // 0..383
  const int mt = tile / 6, np = tile % 6;
  const int m0 = mt * 16, n0 = np * 32;
  const int half = lane >> 4;
  const int l    = lane & 15;

  const int  arow  = m0 + l;                // < 1024
  const int  ncol0 = n0 + l;                // <= 175, always valid
  const int  ncol1 = n0 + 16 + l;           // may be >= 184 (last strip)
  const bool n1ok  = (ncol1 < EMB_N);
  const int  nc1   = n1ok ? ncol1 : 0;
  const float* __restrict__ Ar = x + (size_t)arow * CT_N;

  v8f acc0 = {}, acc1 = {};
  for (int k = 0; k < CT_N; k += 8) {       // 2200 iters, guard-free
    const int ka = k + 2 * half;
    const v2f a0 = *(const v2f*)(Ar + ka);  // 8B-aligned: stride & ka even
    const v2f a1 = *(const v2f*)(Ar + ka + 4);
    v2f b00, b01, b10, b11;
    b00.x = eW[(size_t)ka * EMB_N + ncol0];
    b00.y = eW[(size_t)(ka + 1) * EMB_N + ncol0];
    b01.x = eW[(size_t)(ka + 4) * EMB_N + ncol0];
    b01.y = eW[(size_t)(ka + 5) * EMB_N + ncol0];
    b10.x = eW[(size_t)ka * EMB_N + nc1];
    b10.y = eW[(size_t)(ka + 1) * EMB_N + nc1];
    b11.x = eW[(size_t)(ka + 4) * EMB_N + nc1];
    b11.y = eW[(size_t)(ka + 5) * EMB_N + nc1];
    acc0 = wmma4(a0, b00, acc0);
    acc1 = wmma4(a0, b10, acc1);
    acc0 = wmma4(a1, b01, acc0);
    acc1 = wmma4(a1, b11, acc1);
  }
  const float bn0 = eB[ncol0];
#pragma unroll
  for (int r = 0; r < 8; ++r) {
    const int row = m0 + r + 8 * half;
    h[(size_t)row * EMB_N + ncol0] = acc0[r] + bn0;
  }
  if (n1ok) {
    const float bn1 = eB[ncol1];
#pragma unroll
    for (int r = 0; r < 8; ++r) {
      const int row = m0 + r + 8 * half;
      h[(size_t)row * EMB_N + ncol1] = acc1[r] + bn1;
    }
  }
}

// ---------------------------------------------------------------------------
// Kernel 3: projector layer 1. h1 = h @ W1 + b1 ; g = gelu(h1)
// M=1024 (64 tiles), N=1024 (32 strips, no edges), K=184 (23 iters of 8).
// 2048 waves -> 256 blocks.
// ---------------------------------------------------------------------------
__global__ __launch_bounds__(256) void proj1_gemm_kernel(
    const float* __restrict__ h, const float* __restrict__ W1,
    const float* __restrict__ b1, float* __restrict__ h1,
    float* __restrict__ g) {
  const int wave = threadIdx.x >> 5;
  const int lane = threadIdx.x & 31;
  const int tile = blockIdx.x * 8 + wave;   // 0..2047
  const int mt = tile >> 5, np = tile & 31;
  const int m0 = mt * 16, n0 = np * 32;
  const int half = lane >> 4;
  const int l    = lane & 15;

  const int arow  = m0 + l;
  const int ncol0 = n0 + l;
  const int ncol1 = n0 + 16 + l;            // always < 1024
  const float* __restrict__ Ar = h + (size_t)arow * EMB_N;

  v8f acc0 = {}, acc1 = {};
  for (int k = 0; k < EMB_N; k += 8) {
    const int ka = k + 2 * half;
    const v2f a0 = *(const v2f*)(Ar + ka);
    const v2f a1 = *(const v2f*)(Ar + ka + 4);
    v2f b00, b01, b10, b11;
    b00.x = W1[(size_t)ka * PROJ_N + ncol0];
    b00.y = W1[(size_t)(ka + 1) * PROJ_N + ncol0];
    b01.x = W1[(size_t)(ka + 4) * PROJ_N + ncol0];
    b01.y = W1[(size_t)(ka + 5) * PROJ_N + ncol0];
    b10.x = W1[(size_t)ka * PROJ_N + ncol1];
    b10.y = W1[(size_t)(ka + 1) * PROJ_N + ncol1];
    b11.x = W1[(size_t)(ka + 4) * PROJ_N + ncol1];
    b11.y = W1[(size_t)(ka + 5) * PROJ_N + ncol1];
    acc0 = wmma4(a0, b00, acc0);
    acc1 = wmma4(a0, b10, acc1);
    acc0 = wmma4(a1, b01, acc0);
    acc1 = wmma4(a1, b11, acc1);
  }
  const float bn0 = b1[ncol0];
  const float bn1 = b1[ncol1];
#pragma unroll
  for (int r = 0; r < 8; ++r) {
    const int row = m0 + r + 8 * half;
    const float t0 = acc0[r] + bn0;
    const float t1 = acc1[r] + bn1;
    h1[(size_t)row * PROJ_N + ncol0] = t0;
    h1[(size_t)row * PROJ_N + ncol1] = t1;
    g[(size_t)row * PROJ_N + ncol0] = gelu_exact(t0);
    g[(size_t)row * PROJ_N + ncol1] = gelu_exact(t1);
  }
}

// ---------------------------------------------------------------------------
// Kernel 4: projector layer 2 + residual. hsum = h1 + (g @ W2 + b2)
// M=N=K=1024 (128 iters of 8). 2048 waves -> 256 blocks.
// ---------------------------------------------------------------------------
__global__ __launch_bounds__(256) void proj2_gemm_kernel(
    const float* __restrict__ g, const float* __restrict__ W2,
    const float* __restrict__ b2, const float* __restrict__ h1,
    float* __restrict__ hsum) {
  const int wave = threadIdx.x >> 5;
  const int lane = threadIdx.x & 31;
  const int tile = blockIdx.x * 8 + wave;
  const int mt = tile >> 5, np = tile & 31;
  const int m0 = mt * 16, n0 = np * 32;
  const int half = lane >> 4;
  const int l    = lane & 15;

  const int arow  = m0 + l;
  const int ncol0 = n0 + l;
  const int ncol1 = n0 + 16 + l;
  const float* __restrict__ Ar = g + (size_t)arow * PROJ_N;

  v8f acc0 = {}, acc1 = {};
  for (int k = 0; k < PROJ_N; k += 8) {
    const int ka = k + 2 * half;
    const v2f a0 = *(const v2f*)(Ar + ka);
    const v2f a1 = *(const v2f*)(Ar + ka + 4);
    v2f b00, b01, b10, b11;
    b00.x = W2[(size_t)ka * PROJ_N + ncol0];
    b00.y = W2[(size_t)(ka + 1) * PROJ_N + ncol0];
    b01.x = W2[(size_t)(ka + 4) * PROJ_N + ncol0];
    b01.y = W2[(size_t)(ka + 5) * PROJ_N + ncol0];
    b10.x = W2[(size_t)ka * PROJ_N + ncol1];
    b10.y = W2[(size_t)(ka + 1) * PROJ_N + ncol1];
    b11.x = W2[(size_t)(ka + 4) * PROJ_N + ncol1];
    b11.y = W2[(size_t)(ka + 5) * PROJ_N + ncol1];
    acc0 = wmma4(a0, b00, acc0);
    acc1 = wmma4(a0, b10, acc1);
    acc0 = wmma4(a1, b01, acc0);
    acc1 = wmma4(a1, b11, acc1);
  }
  const float bn0 = b2[ncol0];
  const float bn1 = b2[ncol1];
#pragma unroll
  for (int r = 0; r < 8; ++r) {
    const int row = m0 + r + 8 * half;
    hsum[(size_t)row * PROJ_N + ncol0] =
        h1[(size_t)row * PROJ_N + ncol0] + acc0[r] + bn0;
    hsum[(size_t)row * PROJ_N + ncol1] =
        h1[(size_t)row * PROJ_N + ncol1] + acc1[r] + bn1;
  }
}

// ---------------------------------------------------------------------------
// Kernel 5: LayerNorm over last dim (1024). One block (256 thr) per row.
// ---------------------------------------------------------------------------
__global__ __launch_bounds__(256) void layernorm_kernel(
    const float* __restrict__ hsum, const float* __restrict__ gamma,
    const float* __restrict__ beta, float* __restrict__ out) {
  const int row = blockIdx.x;
  const int tid = threadIdx.x;
  const float* __restrict__ hr = hsum + (size_t)row * PROJ_N;

  float v[4];
  float s = 0.0f, ss = 0.0f;
#pragma unroll
  for (int i = 0; i < 4; ++i) {
    v[i] = hr[tid + i * 256];
    s += v[i];
    ss += v[i] * v[i];
  }
  __shared__ float rs[256];
  __shared__ float rss[256];
  rs[tid] = s;
  rss[tid] = ss;
  __syncthreads();
  for (int off = 128; off > 0; off >>= 1) {
    if (tid < off) {
      rs[tid] += rs[tid + off];
      rss[tid] += rss[tid + off];
    }
    __syncthreads();
  }
  const float mu  = rs[0] * (1.0f / PROJ_N);
  const float var = rss[0] * (1.0f / PROJ_N) - mu * mu;
  const float inv = rsqrtf(var + 1e-5f);
#pragma unroll
  for (int i = 0; i < 4; ++i) {
    const int c = tid + i * 256;
    out[(size_t)row * PROJ_N + c] = (v[i] - mu) * inv * gamma[c] + beta[c];
  }
}

// ---------------------------------------------------------------------------
extern "C" void kernel_launch(void* const* d_in, const int* in_sizes, int n_in,
                              void* d_out, int out_size, void* d_ws,
                              size_t ws_size, hipStream_t stream) {
  const float* eeg    = (const float*)d_in[0];
  const int*   sid    = (const int*)d_in[1];
  const float* subj_W = (const float*)d_in[2];
  const float* subj_b = (const float*)d_in[3];
  const float* enc_W  = (const float*)d_in[4];
  const float* enc_b  = (const float*)d_in[5];
  const float* W1     = (const float*)d_in[6];
  const float* b1     = (const float*)d_in[7];
  const float* W2     = (const float*)d_in[8];
  const float* b2     = (const float*)d_in[9];
  const float* gamma  = (const float*)d_in[10];
  const float* beta   = (const float*)d_in[11];
  float*       out    = (float*)d_out;

  // Workspace layout (floats): ~85.4 MB total
  float* xbuf  = (float*)d_ws;                       // [1024, 17600]
  float* hbuf  = xbuf + (size_t)B_N * CT_N;          // [1024, 184]
  float* h1buf = hbuf + (size_t)B_N * EMB_N;         // [1024, 1024]
  float* gbuf  = h1buf + (size_t)B_N * PROJ_N;       // [1024, 1024]
  float* hsbuf = gbuf + (size_t)B_N * PROJ_N;        // [1024, 1024]

  // 1) subject alignment: 36 strips/batch, 4 waves/block -> 9 blocks x 1024
  subj_gemm_kernel<<<dim3(9, B_N), 128, 0, stream>>>(eeg, sid, subj_W, subj_b,
                                                     xbuf);
  // 2) encoder: 64 x 6 = 384 strips -> 48 blocks
  enc_gemm_kernel<<<48, 256, 0, stream>>>(xbuf, enc_W, enc_b, hbuf);
  // 3) projector L1 + GELU: 64 x 32 = 2048 strips -> 256 blocks
  proj1_gemm_kernel<<<256, 256, 0, stream>>>(hbuf, W1, b1, h1buf, gbuf);
  // 4) projector L2 + residual: 256 blocks
  proj2_gemm_kernel<<<256, 256, 0, stream>>>(gbuf, W2, b2, h1buf, hsbuf);
  // 5) LayerNorm: one block per row
  layernorm_kernel<<<B_N, 256, 0, stream>>>(hsbuf, gamma, beta, out);
}